// PostProcessor_58025008169394
// MI455X (gfx1250) — compile-verified
//
#include <hip/hip_runtime.h>
#include <math.h>

#define NPROP 1500
#define NCLS  81
#define NFG   80
#define NPAD  2048          // next pow2 >= NPROP for bitonic sort
#define SORT_THREADS 256    // 8 waves on wave32
#define TOPK_THREADS 1024
#define DETS  100
#define SCORE_TH 0.05f
#define NMS_TH   0.5f
#define BBOX_CLIP 4.135166556742356f   // log(1000/16)

// ---------------------------------------------------------------------------
// Kernel 1: per-proposal softmax + per-class box decode + clip.
// Writes class-major: scoreMat[c][n] (fg classes c=0..79 <-> label c+1),
//                     boxMat[c][n][4].
// ---------------------------------------------------------------------------
__global__ __launch_bounds__(128)
void k_softmax_decode(const float* __restrict__ logits,     // [1500,81]
                      const float* __restrict__ boxreg,     // [1500,324]
                      const float* __restrict__ props,      // [1500,4]
                      const int*   __restrict__ hPtr,
                      const int*   __restrict__ wPtr,
                      float* __restrict__ scoreMat,         // [80,1500]
                      float* __restrict__ boxMat)           // [80,1500,4]
{
  const int n = blockIdx.x;
  const int t = threadIdx.x;

  __shared__ float  sh_logit[NCLS];
  __shared__ float  sh_max, sh_sum;
  __shared__ float4 sh_prop;

  if (t < NCLS) {
    sh_logit[t] = logits[n * NCLS + t];
    // CDNA5 speculative prefetch of this thread's regression quad
    __builtin_prefetch(&boxreg[n * (4 * NCLS) + 4 * t], 0, 0);
  }
  if (t == 0) sh_prop = ((const float4*)props)[n];
  __syncthreads();

  if (t == 0) {
    float m = sh_logit[0];
    for (int i = 1; i < NCLS; ++i) m = fmaxf(m, sh_logit[i]);
    float s = 0.0f;
    for (int i = 0; i < NCLS; ++i) s += expf(sh_logit[i] - m);
    sh_max = m;
    sh_sum = s;
  }
  __syncthreads();

  if (t >= 1 && t < NCLS) {       // foreground classes only (skip background 0)
    const float prob = expf(sh_logit[t] - sh_max) / sh_sum;

    const float W = (float)wPtr[0];
    const float H = (float)hPtr[0];
    const float w  = sh_prop.z - sh_prop.x + 1.0f;
    const float h  = sh_prop.w - sh_prop.y + 1.0f;
    const float cx = sh_prop.x + 0.5f * w;
    const float cy = sh_prop.y + 0.5f * h;

    const float* r = &boxreg[n * (4 * NCLS) + 4 * t];
    const float dx = r[0] * 0.1f;                       // / wx(10)
    const float dy = r[1] * 0.1f;                       // / wy(10)
    const float dw = fminf(r[2] * 0.2f, BBOX_CLIP);     // / ww(5), clipped
    const float dh = fminf(r[3] * 0.2f, BBOX_CLIP);     // / wh(5), clipped

    const float pcx = dx * w + cx;
    const float pcy = dy * h + cy;
    const float pw  = expf(dw) * w;
    const float ph  = expf(dh) * h;

    float x1 = pcx - 0.5f * pw;
    float y1 = pcy - 0.5f * ph;
    float x2 = pcx + 0.5f * pw - 1.0f;
    float y2 = pcy + 0.5f * ph - 1.0f;

    x1 = fminf(fmaxf(x1, 0.0f), W - 1.0f);
    y1 = fminf(fmaxf(y1, 0.0f), H - 1.0f);
    x2 = fminf(fmaxf(x2, 0.0f), W - 1.0f);
    y2 = fminf(fmaxf(y2, 0.0f), H - 1.0f);

    const int c = t - 1;
    scoreMat[c * NPROP + n] = prob;
    ((float4*)boxMat)[c * NPROP + n] = make_float4(x1, y1, x2, y2);
  }
}

// ---------------------------------------------------------------------------
// Kernel 2: per-class (80 blocks) threshold + descending bitonic sort +
// async-LDS gather of sorted boxes + greedy NMS.
// Emits candidate arrays in sorted-slot order (matches reference flatten).
// ---------------------------------------------------------------------------
__global__ __launch_bounds__(SORT_THREADS)
void k_sort_nms(const float* __restrict__ scoreMat,   // [80,1500]
                const float* __restrict__ boxMat,     // [80,1500,4]
                float* __restrict__ candS,            // [80,1500]
                float* __restrict__ candB)            // [80,1500,4]
{
  const int c   = blockIdx.x;
  const int tid = threadIdx.x;

  __shared__ float  s_key[NPAD];
  __shared__ int    s_idx[NPAD];
  __shared__ float4 s_box[NPROP];
  __shared__ int    s_keep[NPROP];

  // ---- load + threshold mask, pad with -inf ----
  for (int i = tid; i < NPAD; i += SORT_THREADS) {
    float v = -INFINITY;
    if (i < NPROP) {
      const float s = scoreMat[c * NPROP + i];
      v = (s > SCORE_TH) ? s : -INFINITY;
    }
    s_key[i] = v;
    s_idx[i] = i;
  }
  __syncthreads();

  // ---- bitonic sort, descending by key, ties -> smaller idx first ----
  for (unsigned k = 2; k <= NPAD; k <<= 1) {
    for (unsigned j = k >> 1; j > 0; j >>= 1) {
      for (unsigned i = tid; i < NPAD; i += SORT_THREADS) {
        const unsigned p = i ^ j;
        if (p > i) {
          const float a = s_key[i], b = s_key[p];
          const int  ia = s_idx[i], ib = s_idx[p];
          const bool pre  = (a > b) || (a == b && ia < ib);  // a precedes b (desc)
          const bool desc = ((i & k) == 0);
          if (desc != pre) {
            s_key[i] = b; s_key[p] = a;
            s_idx[i] = ib; s_idx[p] = ia;
          }
        }
      }
      __syncthreads();
    }
  }

  // ---- async global->LDS gather of boxes in sorted order (CDNA5 path) ----
  // Per-lane global addr + per-lane LDS dest, 16B per element -> B128.
  {
    const float4* bm = (const float4*)boxMat + (size_t)c * NPROP;
    for (int r = tid; r < NPROP; r += SORT_THREADS) {
      const float4* g = bm + s_idx[r];
      const unsigned lds = (unsigned)(size_t)(&s_box[r]);  // low 32b of flat = LDS offset
      asm volatile("global_load_async_to_lds_b128 %0, %1, off"
                   :
                   : "v"(lds), "v"(g)
                   : "memory");
      s_keep[r] = (s_key[r] > -3.0e38f) ? 1 : 0;           // valid = finite score
    }
    asm volatile("s_wait_asynccnt 0" ::: "memory");        // drain ASYNCcnt
  }
  __syncthreads();

  // ---- greedy NMS (forward elimination == reference greedy keep) ----
  for (int i = 0; i < NPROP - 1; ++i) {
    if (s_keep[i]) {                 // uniform shared read after barrier
      const float4 bi = s_box[i];
      const float  ai = (bi.z - bi.x + 1.0f) * (bi.w - bi.y + 1.0f);
      for (int j = i + 1 + tid; j < NPROP; j += SORT_THREADS) {
        if (s_keep[j]) {
          const float4 bj = s_box[j];
          const float  aj = (bj.z - bj.x + 1.0f) * (bj.w - bj.y + 1.0f);
          const float ix1 = fmaxf(bi.x, bj.x);
          const float iy1 = fmaxf(bi.y, bj.y);
          const float ix2 = fminf(bi.z, bj.z);
          const float iy2 = fminf(bi.w, bj.w);
          const float iw = fmaxf(ix2 - ix1 + 1.0f, 0.0f);
          const float ih = fmaxf(iy2 - iy1 + 1.0f, 0.0f);
          const float inter = iw * ih;
          const float iou = inter / (ai + aj - inter);
          if (iou > NMS_TH) s_keep[j] = 0;
        }
      }
    }
    __syncthreads();
  }

  // ---- emit candidates in sorted-slot order ----
  for (int r = tid; r < NPROP; r += SORT_THREADS) {
    candS[c * NPROP + r] = s_keep[r] ? s_key[r] : -INFINITY;
    ((float4*)candB)[c * NPROP + r] = s_box[r];
  }
}

// ---------------------------------------------------------------------------
// Kernel 3: global top-100 by iterative argmax (smallest index on ties).
// d_out layout: [0:100) scores, [100:500) boxes (100x4), [500:600) labels.
// ---------------------------------------------------------------------------
__global__ __launch_bounds__(TOPK_THREADS)
void k_topk(float* __restrict__ candS,          // [80*1500], mutated (replay-safe)
            const float* __restrict__ candB,    // [80*1500,4]
            float* __restrict__ out)            // 600 floats
{
  __shared__ float rs[TOPK_THREADS];
  __shared__ int   ri[TOPK_THREADS];
  const int tid   = threadIdx.x;
  const int total = NFG * NPROP;

  for (int sel = 0; sel < DETS; ++sel) {
    float best = -INFINITY;
    int   bi   = 0x7fffffff;
    for (int j = tid; j < total; j += TOPK_THREADS) {
      const float v = candS[j];
      if (v > best || (v == best && j < bi)) { best = v; bi = j; }
    }
    rs[tid] = best;
    ri[tid] = bi;
    __syncthreads();
    for (int s = TOPK_THREADS / 2; s > 0; s >>= 1) {
      if (tid < s) {
        const float vo = rs[tid + s];
        const int   io = ri[tid + s];
        if (vo > rs[tid] || (vo == rs[tid] && io < ri[tid])) {
          rs[tid] = vo; ri[tid] = io;
        }
      }
      __syncthreads();
    }
    if (tid == 0) {
      const float s0  = rs[0];
      const int   i0  = ri[0];
      const bool  fin = (s0 > -3.0e38f);
      out[sel] = fin ? s0 : 0.0f;
      float4 b = make_float4(0.0f, 0.0f, 0.0f, 0.0f);
      if (fin) b = ((const float4*)candB)[i0];
      out[DETS + sel * 4 + 0] = b.x;
      out[DETS + sel * 4 + 1] = b.y;
      out[DETS + sel * 4 + 2] = b.z;
      out[DETS + sel * 4 + 3] = b.w;
      out[5 * DETS + sel] = fin ? (float)(i0 / NPROP + 1) : -1.0f;
      if (fin) candS[i0] = -INFINITY;   // remove from next pass
    }
    __syncthreads();                    // also publishes the global write
  }
}

// ---------------------------------------------------------------------------
extern "C" void kernel_launch(void* const* d_in, const int* in_sizes, int n_in,
                              void* d_out, int out_size, void* d_ws, size_t ws_size,
                              hipStream_t stream) {
  const float* logits = (const float*)d_in[0];   // [1500,81]
  const float* boxreg = (const float*)d_in[1];   // [1500,324]
  const float* props  = (const float*)d_in[2];   // [1500,4]
  const int*   hPtr   = (const int*)d_in[3];     // scalar 800
  const int*   wPtr   = (const int*)d_in[4];     // scalar 1333

  // Workspace layout (floats), all 16B-aligned offsets:
  //   scoreMat : 80*1500          = 120000
  //   boxMat   : 80*1500*4        = 480000
  //   candS    : 80*1500          = 120000
  //   candB    : 80*1500*4        = 480000
  // total 1.2M floats = 4.8 MB
  float* ws       = (float*)d_ws;
  float* scoreMat = ws;
  float* boxMat   = scoreMat + NFG * NPROP;
  float* candS    = boxMat   + NFG * NPROP * 4;
  float* candB    = candS    + NFG * NPROP;

  k_softmax_decode<<<NPROP, 128, 0, stream>>>(logits, boxreg, props, hPtr, wPtr,
                                              scoreMat, boxMat);
  k_sort_nms<<<NFG, SORT_THREADS, 0, stream>>>(scoreMat, boxMat, candS, candB);
  k_topk<<<1, TOPK_THREADS, 0, stream>>>(candS, candB, (float*)d_out);
}